// DeepLSTM_36756330120082
// MI455X (gfx1250) — compile-verified
//
#include <hip/hip_runtime.h>
#include <hip/hip_bf16.h>
#include <math.h>

// ---------------------------------------------------------------------------
// DeepLSTM on MI455X (gfx1250), wave32 + v_wmma_f32_16x16x4_f32.
//
// Prep   : pack all weight matrices into WMMA B-fragment order so the hot
//          loops fetch one coalesced b64 per fragment (instead of 2 strided
//          b32). Zero-padding for K=50 baked in.
// Phase 1: sequential LSTM recurrence, 2 workgroups (16-row batch halves),
//          h/c state in LDS, 2 interleaved WMMA accumulator chains per wave.
// Phase 2: fc GEMM [16384,100]x[100,10000]+bias, 16x64 strip per wave
//          (4 interleaved WMMA chains), A tile staged in LDS.
// ---------------------------------------------------------------------------

typedef __attribute__((ext_vector_type(2))) float v2f;
typedef __attribute__((ext_vector_type(8))) float v8f;

__device__ __forceinline__ v8f wmma4(v2f a, v2f b, v8f c) {
  return __builtin_amdgcn_wmma_f32_16x16x4_f32(false, a, false, b,
                                               (short)0, c, false, false);
}

__device__ __forceinline__ float sigmoidf_(float x) {
  return 1.0f / (1.0f + __expf(-x));
}

__device__ __forceinline__ v8f splat8(float x) {
  v8f r;
#pragma unroll
  for (int v = 0; v < 8; ++v) r[v] = x;
  return r;
}

// ---------------------------------------------------------------------------
// Pack W[Kreal][G] (row-major) into B-fragment order:
//   dst[((nt*ksteps + ks)*32 + lane)*2 + j] =
//       W[ks*4 + (lane>>4)*2 + j][nt*16 + (lane&15)]   (0 if k-row OOB)
// ---------------------------------------------------------------------------
__global__ __launch_bounds__(256)
void pack_b_kernel(const float* __restrict__ src, float* __restrict__ dst,
                   int Kreal, int ksteps, int G) {
  int idx = blockIdx.x * 256 + threadIdx.x;
  int total = (G / 16) * ksteps * 64;
  if (idx >= total) return;
  int j    = idx & 1;
  int lane = (idx >> 1) & 31;
  int rest = idx >> 6;
  int ks   = rest % ksteps;
  int nt   = rest / ksteps;
  int krow = ks * 4 + ((lane >> 4) << 1) + j;
  int col  = nt * 16 + (lane & 15);
  dst[idx] = (krow < Kreal) ? src[(size_t)krow * G + col] : 0.0f;
}

// ---------------------------------------------------------------------------
// Phase 1: LSTM recurrence for one 16-row batch slice.
// ---------------------------------------------------------------------------
__global__ __launch_bounds__(256)
void lstm_seq_kernel(const int*   __restrict__ tokens, // [B,T]
                     const float* __restrict__ emb,    // [V,E]
                     const float* __restrict__ bias0,  // [4H]
                     const float* __restrict__ biasl,  // [L-1,4H]
                     const float* __restrict__ pWx0,   // packed [25][13][64]
                     const float* __restrict__ pWh0,   // packed [25][25][64]
                     const float* __restrict__ pWxL,   // packed 4x[25][25][64]
                     const float* __restrict__ pWhL,   // packed 4x[25][25][64]
                     float*       __restrict__ Hall)   // [B*T,H], row = b*T+t
{
  constexpr int T = 512, E = 50, H = 100, G = 400, NL = 5, MB = 16;
  constexpr int XP = 52;             // E padded to multiple of 4
  constexpr int LSTEP = 25 * 25 * 64; // packed floats per recurrent matrix

  __shared__ float sh[NL][MB][H];
  __shared__ float sc[NL][MB][H];
  __shared__ float sx[MB][XP];
  __shared__ float sg[MB][G];

  const int tid   = threadIdx.x;
  const int wave  = tid >> 5;
  const int lane  = tid & 31;
  const int b0r   = blockIdx.x * MB;
  const int row   = lane & 15;
  const int khalf = (lane >> 4) * 2;
  const int rbase = (lane >> 4) * 8;

  for (int i = tid; i < NL * MB * H; i += 256) {
    (&sh[0][0][0])[i] = 0.0f;
    (&sc[0][0][0])[i] = 0.0f;
  }
  if (tid < MB * 2) sx[tid & 15][E + (tid >> 4)] = 0.0f;
  __syncthreads();

  for (int t = 0; t < T; ++t) {
    // embedding gather -> LDS [16,50]
    for (int i = tid; i < MB * E; i += 256) {
      int r = i / E, e = i - r * E;
      int tok = tokens[(b0r + r) * T + t];
      sx[r][e] = emb[tok * E + e];
    }
    __syncthreads();

    for (int l = 0; l < NL; ++l) {
      const float* Ain;  int Astride, Kxsteps;
      const v2f *pWxV, *pWhV;  const float* bl;
      if (l == 0) {
        Ain = &sx[0][0]; Astride = XP; Kxsteps = 13;
        pWxV = (const v2f*)pWx0;
        pWhV = (const v2f*)pWh0;
        bl   = bias0;
      } else {
        Ain = &sh[l - 1][0][0]; Astride = H; Kxsteps = 25;
        pWxV = (const v2f*)(pWxL + (size_t)(l - 1) * LSTEP);
        pWhV = (const v2f*)(pWhL + (size_t)(l - 1) * LSTEP);
        bl   = biasl + (size_t)(l - 1) * G;
      }
      const float* Hprev = &sh[l][0][0];

      // gates[16,400] = bias + x@Wx + h@Wh ; two interleaved tiles per wave
      for (int base = wave; base < 25; base += 16) {   // wave-uniform
        const int  ntA  = base;
        const int  ntB  = base + 8;
        const bool hasB = (ntB < 25);
        const int  ntBc = hasB ? ntB : ntA;   // clamp: redundant, store-guarded
        const int  nA   = ntA * 16 + row;
        const int  nB   = ntBc * 16 + row;

        v8f acc0 = splat8(bl[nA]);
        v8f acc1 = splat8(bl[nB]);

        const v2f* BxA = pWxV + (size_t)ntA  * Kxsteps * 32 + lane;
        const v2f* BxB = pWxV + (size_t)ntBc * Kxsteps * 32 + lane;
        for (int ks = 0; ks < Kxsteps; ++ks) {
          v2f a = *(const v2f*)&Ain[row * Astride + ks * 4 + khalf];
          acc0 = wmma4(a, BxA[ks * 32], acc0);
          acc1 = wmma4(a, BxB[ks * 32], acc1);
        }
        const v2f* BhA = pWhV + (size_t)ntA  * 25 * 32 + lane;
        const v2f* BhB = pWhV + (size_t)ntBc * 25 * 32 + lane;
        for (int ks = 0; ks < 25; ++ks) {
          v2f a = *(const v2f*)&Hprev[row * H + ks * 4 + khalf];
          acc0 = wmma4(a, BhA[ks * 32], acc0);
          acc1 = wmma4(a, BhB[ks * 32], acc1);
        }
#pragma unroll
        for (int v = 0; v < 8; ++v) sg[v + rbase][nA] = acc0[v];
        if (hasB) {
#pragma unroll
          for (int v = 0; v < 8; ++v) sg[v + rbase][ntB * 16 + row] = acc1[v];
        }
      }
      __syncthreads();

      // elementwise cell update (gate order i, f, o, c~); bias already in sg
      for (int i = tid; i < MB * H; i += 256) {
        int r = i / H, j = i - r * H;
        float gi = sg[r][j];
        float gf = sg[r][H + j];
        float go = sg[r][2 * H + j];
        float gc = sg[r][3 * H + j];
        float cn = sigmoidf_(gf) * sc[l][r][j] + sigmoidf_(gi) * tanhf(gc);
        float hn = sigmoidf_(go) * tanhf(cn);
        sc[l][r][j] = cn;
        sh[l][r][j] = hn;
        if (l == NL - 1)
          Hall[((size_t)(b0r + r) * T + t) * H + j] = hn;
      }
      __syncthreads();
    }
  }
}

// ---------------------------------------------------------------------------
// Phase 2: out[16384,10000] = Hall @ fc_w + fc_b.
// Each wave: one 16x64 strip = 4 interleaved WMMA chains sharing the A frag.
// grid = (1024 m-tiles, 20 strip-groups), block = 256 (8 waves).
// ---------------------------------------------------------------------------
__global__ __launch_bounds__(256)
void fc_gemm_kernel(const float* __restrict__ Hall,  // [16384,100]
                    const float* __restrict__ pFcw,  // packed [625][25][64]
                    const float* __restrict__ fcb,   // [10000]
                    float*       __restrict__ out)   // [16384,10000]
{
  constexpr int K = 100, N = 10000, NTILES = N / 16; // 625
  __shared__ float sA[16][K];

  const int tid  = threadIdx.x;
  const int wave = tid >> 5;
  const int lane = tid & 31;
  const int m    = blockIdx.x;

  for (int i = tid; i < 16 * K; i += 256) {
    int r = i / K, k = i - r * K;
    sA[r][k] = Hall[(size_t)(m * 16 + r) * K + k];
  }
  __syncthreads();

  const int nt0 = (blockIdx.y * 8 + wave) * 4;
  if (nt0 >= NTILES) return;            // whole-wave exit, EXEC stays full

  const int row   = lane & 15;
  const int khalf = (lane >> 4) * 2;
  const int rbase = (lane >> 4) * 8;

  int nt[4];
  const v2f* Bp[4];
  v8f acc[4];
#pragma unroll
  for (int j = 0; j < 4; ++j) {
    nt[j]  = (nt0 + j < NTILES) ? (nt0 + j) : (NTILES - 1); // clamp
    Bp[j]  = (const v2f*)pFcw + (size_t)nt[j] * 25 * 32 + lane;
    acc[j] = splat8(fcb[nt[j] * 16 + row]);  // bias folded into C
  }

  for (int ks = 0; ks < 25; ++ks) {
    v2f a = *(const v2f*)&sA[row][ks * 4 + khalf];
#pragma unroll
    for (int j = 0; j < 4; ++j) acc[j] = wmma4(a, Bp[j][ks * 32], acc[j]);
  }

#pragma unroll
  for (int j = 0; j < 4; ++j) {
    if (nt0 + j < NTILES) {
      const int n = nt[j] * 16 + row;
#pragma unroll
      for (int v = 0; v < 8; ++v)
        out[(size_t)(m * 16 + v + rbase) * N + n] = acc[j][v];
    }
  }
}

// ---------------------------------------------------------------------------
extern "C" void kernel_launch(void* const* d_in, const int* in_sizes, int n_in,
                              void* d_out, int out_size, void* d_ws, size_t ws_size,
                              hipStream_t stream) {
  const int*   tokens = (const int*)  d_in[0]; // [32,512]
  const float* emb    = (const float*)d_in[1]; // [10000,50]
  const float* Wx0    = (const float*)d_in[2]; // [50,400]
  const float* Wh0    = (const float*)d_in[3]; // [100,400]
  const float* b0     = (const float*)d_in[4]; // [400]
  const float* Wx     = (const float*)d_in[5]; // [4,100,400]
  const float* Wh     = (const float*)d_in[6]; // [4,100,400]
  const float* bl     = (const float*)d_in[7]; // [4,400]
  const float* fcw    = (const float*)d_in[8]; // [100,10000]
  const float* fcb    = (const float*)d_in[9]; // [10000]
  float*       out    = (float*)d_out;         // [32,512,10000]

  // workspace layout (floats)
  float* ws    = (float*)d_ws;
  float* Hall  = ws;                           // 16384*100      = 1,638,400
  float* pWx0  = Hall  + 1638400;              // 25*13*64       =    20,800
  float* pWh0  = pWx0  + 20800;                // 25*25*64       =    40,000
  float* pWxL  = pWh0  + 40000;                // 4*40,000       =   160,000
  float* pWhL  = pWxL  + 160000;               // 4*40,000       =   160,000
  float* pFcw  = pWhL  + 160000;               // 625*25*64      = 1,000,000
                                               // total 3,019,200 f = 12.1 MB

  // ---- prep: pack weights into B-fragment order (all L2-resident) ----
  pack_b_kernel<<<dim3((25 * 13 * 64 + 255) / 256), 256, 0, stream>>>(
      Wx0, pWx0, 50, 13, 400);
  pack_b_kernel<<<dim3((25 * 25 * 64 + 255) / 256), 256, 0, stream>>>(
      Wh0, pWh0, 100, 25, 400);
  for (int l = 0; l < 4; ++l) {
    pack_b_kernel<<<dim3((25 * 25 * 64 + 255) / 256), 256, 0, stream>>>(
        Wx + (size_t)l * 100 * 400, pWxL + (size_t)l * 40000, 100, 25, 400);
    pack_b_kernel<<<dim3((25 * 25 * 64 + 255) / 256), 256, 0, stream>>>(
        Wh + (size_t)l * 100 * 400, pWhL + (size_t)l * 40000, 100, 25, 400);
  }
  pack_b_kernel<<<dim3((625 * 25 * 64 + 255) / 256), 256, 0, stream>>>(
      fcw, pFcw, 100, 25, 10000);

  // ---- phase 1: two independent 16-row recurrences ----
  lstm_seq_kernel<<<dim3(2), dim3(256), 0, stream>>>(
      tokens, emb, b0, bl, pWx0, pWh0, pWxL, pWhL, Hall);

  // ---- phase 2: big fc GEMM ----
  fc_gemm_kernel<<<dim3(1024, 20), dim3(256), 0, stream>>>(Hall, pFcw, fcb, out);
}